// baselineGRU_26139170963944
// MI455X (gfx1250) — compile-verified
//
#include <hip/hip_runtime.h>
#include <cmath>

typedef __attribute__((ext_vector_type(2))) float v2f;
typedef __attribute__((ext_vector_type(8))) float v8f;
typedef __attribute__((ext_vector_type(4))) unsigned int u32x4;
typedef __attribute__((ext_vector_type(8))) int i32x8;
typedef __attribute__((ext_vector_type(4))) int i32x4;

#define B_   64
#define S_   2048
#define IN_  208
#define H_   100
#define G3_  300
#define OUT_ 98

__device__ __forceinline__ float sigmoidf_(float x) {
  return 1.0f / (1.0f + __expf(-x));
}

// ---------------------------------------------------------------------------
// Generic fp32 WMMA GEMM:  C[M,Nreal] = A[M,K] * W[Nreal,K]^T  (+ bias)
// One wave per 16x16 output tile, K accumulated with V_WMMA_F32_16X16X4_F32.
// ---------------------------------------------------------------------------
__launch_bounds__(256, 1)
__global__ void wmma_gemm_f32(const float* __restrict__ A,
                              const float* __restrict__ W,
                              const float* __restrict__ bias,
                              float* __restrict__ C,
                              int M, int Nreal, int K)
{
  const int  ntiles = (Nreal + 15) >> 4;
  const long tileId = (long)blockIdx.x * 8 + (threadIdx.x >> 5);
  const long mt = tileId / ntiles;
  const int  nt = (int)(tileId % ntiles);
  if (mt * 16 >= (long)M) return;           // wave-uniform exit

  const int lane    = threadIdx.x & 31;
  const int halfsel = lane >> 4;            // 0: lanes 0-15, 1: lanes 16-31
  const int lm      = lane & 15;
  const int koff    = halfsel * 2;          // A/B: second half-wave holds k+2,k+3
  const int n       = nt * 16 + lm;
  const int nld     = (n < Nreal) ? n : (Nreal - 1);   // clamp OOB loads

  const float* arow = A + ((long)mt * 16 + lm) * K;
  const float* wrow = W + (long)nld * K;

  v8f acc = {};
  for (int kb = 0; kb < K; kb += 4) {
    v2f a = *(const v2f*)(arow + kb + koff);
    v2f b = *(const v2f*)(wrow + kb + koff);
    acc = __builtin_amdgcn_wmma_f32_16x16x4_f32(false, a, false, b,
                                                (short)0, acc, false, false);
  }

  if (n < Nreal) {
    const float bv = bias ? bias[n] : 0.0f;
    const long  rbase = mt * 16 + (halfsel << 3);   // C/D: rows v + 8*halfsel
    #pragma unroll
    for (int v = 0; v < 8; ++v)
      C[(rbase + v) * Nreal + n] = acc[v] + bv;
  }
}

// ---------------------------------------------------------------------------
// TDM helper: async-copy a 16-row x 300-col fp32 tile (row stride S*3H elems)
// from global memory into LDS. D# per CDNA5 ISA §8.3/8.4; 2D tensor so the
// trailing descriptor groups are zero. Tracked by TENSORcnt.
// amdgpu-toolchain (clang-23) 6-arg builtin form.
// ---------------------------------------------------------------------------
__device__ __forceinline__ void tdm_load_xg_tile(const float* gsrc, unsigned lds_off)
{
  unsigned long long ga = (unsigned long long)(uintptr_t)gsrc;
  u32x4 g0;
  g0[0] = 1u;                                          // count=1, user mode
  g0[1] = lds_off;                                     // lds_addr (bytes)
  g0[2] = (unsigned)(ga & 0xFFFFFFFFu);                // global_addr[31:0]
  g0[3] = (unsigned)((ga >> 32) & 0x01FFFFFFu)         // global_addr[56:32]
        | (2u << 30);                                  // type=2 ("image")
  i32x8 g1;
  g1[0] = 0x00020000;                  // workgroup_mask=0, data_size=2 (4B)
  g1[1] = (int)(((unsigned)G3_ & 0xFFFFu) << 16);      // tensor_dim0 lo = 300
  g1[2] = (int)(16u << 16);            // tensor_dim0 hi=0, tensor_dim1 lo=16
  g1[3] = (int)((unsigned)G3_ << 16);  // tensor_dim1 hi=0, tile_dim0=300
  g1[4] = 16;                          // tile_dim1=16, tile_dim2=0
  g1[5] = (int)(S_ * G3_);             // tensor_dim0_stride lo (elements)
  g1[6] = 0;                           // stride hi=0, tensor_dim1_stride lo=0
  g1[7] = 0;
  i32x4 z4 = {0, 0, 0, 0};
  i32x8 z8 = {0, 0, 0, 0, 0, 0, 0, 0};
  __builtin_amdgcn_tensor_load_to_lds(g0, g1, z4, z4, z8, 0);
}

// ---------------------------------------------------------------------------
// Persistent GRU recurrence. 4 workgroups (16 batch rows each), 256 threads.
// W_hh resident in LDS; hg = h @ W_hh^T via fp32 WMMA each step; xg gate
// pre-activations double-buffered in LDS and prefetched one step ahead by
// the Tensor Data Mover so phase B never touches global memory for reads.
// ---------------------------------------------------------------------------
__launch_bounds__(256, 1)
__global__ void gru_recurrence(const float* __restrict__ xg,   // [B,S,3H] no bias
                               const float* __restrict__ h0,   // [B,H]
                               const float* __restrict__ w_hh, // [3H,H]
                               const float* __restrict__ b_ih,
                               const float* __restrict__ b_hh,
                               float* __restrict__ y)          // [B,S,H]
{
  __shared__ float wlds[304][104];     // W_hh rows (n-major, k fast), padded
  __shared__ float hbuf[16][104];      // hidden state, 16 batch rows
  __shared__ float hg[16][304];        // per-step h @ W_hh^T
  __shared__ float xtile[2][16][G3_];  // TDM double buffer of xg tiles
  __shared__ float bsum[304];          // b_ih + b_hh (r,z gates)
  __shared__ float bihn[104];          // b_ih n-gate
  __shared__ float bhhn[104];          // b_hh n-gate

  const int tid     = threadIdx.x;
  const int lane    = tid & 31;
  const int wave    = tid >> 5;
  const int lm      = lane & 15;
  const int halfsel = lane >> 4;
  const int koff    = halfsel * 2;
  const int bbase   = blockIdx.x * 16;

  const unsigned xoff[2] = { (unsigned)(uintptr_t)&xtile[0][0][0],
                             (unsigned)(uintptr_t)&xtile[1][0][0] };
  const float* xg_wg = xg + (size_t)bbase * S_ * G3_;   // row 0 of our tile

  for (int idx = tid; idx < G3_ * H_; idx += 256)
    wlds[idx / H_][idx % H_] = w_hh[idx];
  for (int idx = tid; idx < 4 * H_; idx += 256)         // pad rows 300..303
    wlds[G3_ + idx / H_][idx % H_] = 0.0f;
  for (int g = tid; g < G3_; g += 256)
    bsum[g] = b_ih[g] + b_hh[g];
  for (int k = tid; k < H_; k += 256) {
    bihn[k] = b_ih[2 * H_ + k];
    bhhn[k] = b_hh[2 * H_ + k];
  }
  for (int idx = tid; idx < 16 * H_; idx += 256)
    hbuf[idx / H_][idx % H_] = h0[(bbase + idx / H_) * H_ + (idx % H_)];

  // Preload xg tile for t=0 via TDM (wave 0 issues & drains its TENSORcnt).
  if (wave == 0) {
    tdm_load_xg_tile(xg_wg, xoff[0]);
    __builtin_amdgcn_s_wait_tensorcnt(0);
  }
  __syncthreads();

  // Tile ownership: wave w handles N-tiles {w, w+8, w+16(<19)}
  const int j0 = wave, j1 = wave + 8, j2 = wave + 16;
  int cur = 0;

  for (int t = 0; t < S_; ++t) {
    // Kick off TDM prefetch of step t+1's xg tile into the other buffer.
    if (wave == 0 && t + 1 < S_)
      tdm_load_xg_tile(xg_wg + (size_t)(t + 1) * G3_, xoff[cur ^ 1]);

    // --- Phase A: hg = h @ W_hh^T (fp32 WMMA, K=100 = 25 x 4) ---
    v8f acc0 = {}, acc1 = {}, acc2 = {};
    for (int ks = 0; ks < 25; ++ks) {
      const int kb = ks * 4 + koff;
      v2f a = *(const v2f*)&hbuf[lm][kb];
      {
        v2f b = *(const v2f*)&wlds[j0 * 16 + lm][kb];
        acc0 = __builtin_amdgcn_wmma_f32_16x16x4_f32(false, a, false, b,
                                                     (short)0, acc0, false, false);
      }
      {
        v2f b = *(const v2f*)&wlds[j1 * 16 + lm][kb];
        acc1 = __builtin_amdgcn_wmma_f32_16x16x4_f32(false, a, false, b,
                                                     (short)0, acc1, false, false);
      }
      if (j2 < 19) {  // wave-uniform
        v2f b = *(const v2f*)&wlds[j2 * 16 + lm][kb];
        acc2 = __builtin_amdgcn_wmma_f32_16x16x4_f32(false, a, false, b,
                                                     (short)0, acc2, false, false);
      }
    }
    const int rb = halfsel << 3;
    #pragma unroll
    for (int v = 0; v < 8; ++v) {
      hg[rb + v][j0 * 16 + lm] = acc0[v];
      hg[rb + v][j1 * 16 + lm] = acc1[v];
    }
    if (j2 < 19) {
      #pragma unroll
      for (int v = 0; v < 8; ++v)
        hg[rb + v][j2 * 16 + lm] = acc2[v];
    }
    __syncthreads();

    // --- Phase B: gates + state update, all operands from LDS ---
    for (int idx = tid; idx < 16 * H_; idx += 256) {
      const int r = idx / H_, k = idx % H_;
      const float rr = sigmoidf_(xtile[cur][r][k]          + hg[r][k]          + bsum[k]);
      const float zz = sigmoidf_(xtile[cur][r][H_ + k]     + hg[r][H_ + k]     + bsum[H_ + k]);
      const float nn = tanhf(xtile[cur][r][2 * H_ + k] + bihn[k] +
                             rr * (hg[r][2 * H_ + k] + bhhn[k]));
      const float hnew = (1.0f - zz) * nn + zz * hbuf[r][k];
      hbuf[r][k] = hnew;
      y[(((long)(bbase + r)) * S_ + t) * H_ + k] = hnew;
    }

    // Drain the prefetch before anyone can consume the next buffer.
    if (wave == 0)
      __builtin_amdgcn_s_wait_tensorcnt(0);
    __syncthreads();
    cur ^= 1;
  }
}

// ---------------------------------------------------------------------------
extern "C" void kernel_launch(void* const* d_in, const int* in_sizes, int n_in,
                              void* d_out, int out_size, void* d_ws, size_t ws_size,
                              hipStream_t stream)
{
  const float* seq     = (const float*)d_in[0];
  const float* h0      = (const float*)d_in[1];
  const float* w_ih_l0 = (const float*)d_in[2];
  const float* w_hh_l0 = (const float*)d_in[3];
  const float* b_ih_l0 = (const float*)d_in[4];
  const float* b_hh_l0 = (const float*)d_in[5];
  const float* w_ih_l1 = (const float*)d_in[6];
  const float* w_hh_l1 = (const float*)d_in[7];
  const float* b_ih_l1 = (const float*)d_in[8];
  const float* b_hh_l1 = (const float*)d_in[9];
  const float* w_out   = (const float*)d_in[10];
  const float* b_out   = (const float*)d_in[11];
  float* out = (float*)d_out;

  const int M = B_ * S_;                        // 131072 rows
  float* xg   = (float*)d_ws;                   // [M,300]  157.3 MB
  float* ybuf = xg + (size_t)M * G3_;           // [M,100]   52.4 MB (reused)

  const long gtiles3 = (long)(M / 16) * ((G3_ + 15) / 16);
  const long gtilesO = (long)(M / 16) * ((OUT_ + 15) / 16);
  const int  blk3    = (int)((gtiles3 + 7) / 8);
  const int  blkO    = (int)((gtilesO + 7) / 8);

  // Layer 0: xg0 = seq @ W_ih0^T  (biases folded into recurrence)
  wmma_gemm_f32<<<blk3, 256, 0, stream>>>(seq, w_ih_l0, nullptr, xg, M, G3_, IN_);
  gru_recurrence<<<4, 256, 0, stream>>>(xg, h0, w_hh_l0, b_ih_l0, b_hh_l0, ybuf);

  // Layer 1: xg1 = y0 @ W_ih1^T (overwrites xg region; y1 overwrites y0 region)
  wmma_gemm_f32<<<blk3, 256, 0, stream>>>(ybuf, w_ih_l1, nullptr, xg, M, G3_, H_);
  gru_recurrence<<<4, 256, 0, stream>>>(xg, h0 + (size_t)B_ * H_,
                                        w_hh_l1, b_ih_l1, b_hh_l1, ybuf);

  // Output projection: out = y1 @ W_out^T + b_out
  wmma_gemm_f32<<<blkO, 256, 0, stream>>>(ybuf, w_out, b_out, out, M, OUT_, H_);

  (void)in_sizes; (void)n_in; (void)out_size; (void)ws_size;
}